// E2N4DLayer_2190433321020
// MI455X (gfx1250) — compile-verified
//
#include <hip/hip_runtime.h>

typedef __attribute__((ext_vector_type(4))) float v4f;
typedef __attribute__((ext_vector_type(2))) float v2f;
typedef __attribute__((ext_vector_type(8))) float v8f;

constexpr int kC    = 32;
constexpr int kMin  = 16;
constexpr int kMout = 32;
constexpr int kB    = 4;
constexpr int kNCube = kB * kMin * kC;  // 2048 cubes of 32^3 floats

// ---------------------------------------------------------------------------
// Phase 1: streaming reduction. One block per (b, m, c2) cube (32768 floats,
// contiguous). float4 loads => c3 == iteration index, c4 == t>>3 (fixed per
// thread), c5 collapsed inside the float4. NT loads: data is touched exactly
// once, keep it out of L2's way. No global atomics -> deterministic.
// ---------------------------------------------------------------------------
__global__ __launch_bounds__(256) void phase1_reduce(const float* __restrict__ x,
                                                     float* __restrict__ p3,
                                                     float* __restrict__ p4) {
  __shared__ float s3_partial[8][kC];   // per-wave, per-c3 partials
  __shared__ float s4_final[kC];        // wave w exclusively owns c4 in [4w,4w+4)

  const int t = threadIdx.x;
  const int w = t >> 5;
  const int l = t & 31;
  const int cube = blockIdx.x;

  const v4f* __restrict__ xc = (const v4f*)(x + (size_t)cube * (kC * kC * kC));

  float acc4 = 0.f;  // accumulates this thread's fixed c4 = t>>3
  for (int i = 0; i < kC; ++i) {        // i == c3
    v4f v = __builtin_nontemporal_load(&xc[i * 256 + t]);
    float s = (v.x + v.y) + (v.z + v.w);
    acc4 += s;
    // block needs sum over all 256 threads for s3[c3=i]; do wave reduce here
    s += __shfl_xor(s, 16, 32);
    s += __shfl_xor(s, 8, 32);
    s += __shfl_xor(s, 4, 32);
    s += __shfl_xor(s, 2, 32);
    s += __shfl_xor(s, 1, 32);
    if (l == 0) s3_partial[w][i] = s;
  }
  // reduce acc4 across the 8 lanes sharing c4 (lanes l, l^1, l^2, l^4)
  acc4 += __shfl_xor(acc4, 1, 32);
  acc4 += __shfl_xor(acc4, 2, 32);
  acc4 += __shfl_xor(acc4, 4, 32);
  if ((l & 7) == 0) s4_final[4 * w + (l >> 3)] = acc4;
  __syncthreads();

  if (t < kC) {                         // fold 8 wave partials, write p3
    float r = 0.f;
#pragma unroll
    for (int w2 = 0; w2 < 8; ++w2) r += s3_partial[w2][t];
    p3[cube * kC + t] = r;
  } else if (t < 2 * kC) {              // write p4
    p4[cube * kC + (t - kC)] = s4_final[t - kC];
  }
}

// ---------------------------------------------------------------------------
// Phase 2: fold partials over c2 into S[b][m][i][c] (LDS), sum the 4 weight
// mats into W[m][n], then out(b,i) = W^T(32x16) @ S(16x32) via 4 chained
// V_WMMA_F32_16X16X4_F32 per 16x16 tile. 8 waves x 8 tiles = 64 tiles.
// ---------------------------------------------------------------------------
__global__ __launch_bounds__(256) void phase2_wmma(const float* __restrict__ p3,
                                                   const float* __restrict__ p4,
                                                   const float* __restrict__ alpha,
                                                   const float* __restrict__ beta,
                                                   const float* __restrict__ gamma_,
                                                   const float* __restrict__ delta,
                                                   float* __restrict__ out) {
  __shared__ float S[kB][kMin][4][kC];   // 32 KB
  __shared__ float W[kMin][kMout];       // 2 KB

  const int t = threadIdx.x;

  // ---- Step A: s1 (== s2), s3, s4 per (b,m,c) ----
  {
    const int bm = t >> 2;   // 0..63
    const int q  = t & 3;
    const int b = bm >> 4, m = bm & 15;
#pragma unroll
    for (int cc = 0; cc < 8; ++cc) {
      const int c = q * 8 + cc;
      float s1 = 0.f, s3 = 0.f, s4 = 0.f;
      for (int k = 0; k < kC; ++k) {
        s1 += p3[(bm * kC + c) * kC + k];  // sum over c3 for cube (bm,c2=c)
        s3 += p3[(bm * kC + k) * kC + c];  // sum over c2, fixed c3=c
        s4 += p4[(bm * kC + k) * kC + c];  // sum over c2, fixed c4=c
      }
      S[b][m][0][c] = s1;
      S[b][m][1][c] = s1;
      S[b][m][2][c] = s3;
      S[b][m][3][c] = s4;
    }
  }

  // ---- Step B: W = alpha + beta + gamma + delta ----
  for (int e = t; e < kMin * kMout; e += 256)
    W[e >> 5][e & 31] = alpha[e] + beta[e] + gamma_[e] + delta[e];

  __syncthreads();

  // ---- Step C: WMMA tiles ----
  const int w  = t >> 5;
  const int l  = t & 31;
  const int lh = l >> 4;   // lane half selects K+2 (A/B) and M+8 (D)
  const int ll = l & 15;

  for (int j = 0; j < 8; ++j) {
    const int tau = w * 8 + j;           // 64 output tiles
    const int b  = tau >> 4;
    const int i  = (tau >> 2) & 3;
    const int tn = (tau >> 1) & 1;
    const int tc = tau & 1;

    v8f acc = {};
#pragma unroll
    for (int k0 = 0; k0 < kMin; k0 += 4) {
      // A (16x4, M=n_local, K=m): lane=M%16, vgpr v + 2*half = K
      const int m0 = k0 + 0 + 2 * lh;
      const int m1 = k0 + 1 + 2 * lh;
      v2f a, bmat;
      a.x = W[m0][tn * 16 + ll];         // A[n][m] = W^T
      a.y = W[m1][tn * 16 + ll];
      // B (4x16, K=m, N=c): vgpr v + 2*half = K, lane = N
      bmat.x = S[b][m0][i][tc * 16 + ll];
      bmat.y = S[b][m1][i][tc * 16 + ll];
      acc = __builtin_amdgcn_wmma_f32_16x16x4_f32(
          /*neg_a=*/false, a, /*neg_b=*/false, bmat,
          /*c_mod=*/(short)0, acc, /*reuse_a=*/false, /*reuse_b=*/false);
    }
    // D layout: vgpr v, lane half -> row n_local = v + 8*lh; lane%16 -> col
#pragma unroll
    for (int v = 0; v < 8; ++v) {
      const int n = tn * 16 + v + 8 * lh;
      const int c = tc * 16 + ll;
      out[((b * kMout + n) * 4 + i) * kC + c] = acc[v];
    }
  }
}

// ---------------------------------------------------------------------------
extern "C" void kernel_launch(void* const* d_in, const int* in_sizes, int n_in,
                              void* d_out, int out_size, void* d_ws, size_t ws_size,
                              hipStream_t stream) {
  const float* x      = (const float*)d_in[0];
  const float* alpha  = (const float*)d_in[1];
  const float* beta   = (const float*)d_in[2];
  const float* gamma_ = (const float*)d_in[3];
  const float* delta  = (const float*)d_in[4];
  float* out = (float*)d_out;

  float* p3 = (float*)d_ws;                 // [2048][32]
  float* p4 = p3 + (size_t)kNCube * kC;     // [2048][32]  (512 KB total)

  phase1_reduce<<<kNCube, 256, 0, stream>>>(x, p3, p4);
  phase2_wmma<<<1, 256, 0, stream>>>(p3, p4, alpha, beta, gamma_, delta, out);
}